// SparseGrid_42511586296075
// MI455X (gfx1250) — compile-verified
//
#include <hip/hip_runtime.h>

typedef float f4 __attribute__((ext_vector_type(4)));

#define RESO     192
#define DATA_F4  7          // 28 floats = 7 x float4 per voxel row
#define BLOCK    256

__global__ __launch_bounds__(BLOCK) void trilerp_sparse_kernel(
    const float* __restrict__ points,
    const f4*    __restrict__ data,
    const int*   __restrict__ links,
    const float* __restrict__ offs,
    const float* __restrict__ scal,
    f4*          __restrict__ out,
    int npts)
{
    __shared__ float spts[BLOCK * 3];
    const int tid        = (int)threadIdx.x;
    const int blockStart = (int)blockIdx.x * BLOCK;

    // ---- Stage this block's point coords into LDS via gfx1250 async-to-LDS ----
    // Coalesced: lane j-th pass loads float (blockStart*3 + tid + j*256).
    {
        const int lastE = npts * 3 - 1;
#pragma unroll
        for (int j = 0; j < 3; ++j) {
            int le = tid + j * BLOCK;            // element inside block tile
            int ge = blockStart * 3 + le;        // global float index
            if (ge > lastE) ge = lastE;          // tail clamp (dup reads, unused)
            unsigned ldsAddr = (unsigned)(unsigned long long)(&spts[le]);
            unsigned gOff    = (unsigned)ge * 4u;
            asm volatile("global_load_async_to_lds_b32 %0, %1, %2"
                         :: "v"(ldsAddr), "v"(gOff), "s"(points)
                         : "memory");
        }
        asm volatile("s_wait_asynccnt 0" ::: "memory");
        __syncthreads();                         // cross-wave LDS visibility
    }

    const int i = blockStart + tid;
    if (i >= npts) return;

    const float gsz = (float)RESO;

    float p[3];
#pragma unroll
    for (int a = 0; a < 3; ++a) {
        float v = spts[tid * 3 + a] * (scal[a] * gsz) + (offs[a] * gsz - 0.5f);
        v = fminf(fmaxf(v, 0.0f), gsz - 1.0f);
        p[a] = v;
    }
    int lx = (int)p[0]; if (lx > RESO - 2) lx = RESO - 2;
    int ly = (int)p[1]; if (ly > RESO - 2) ly = RESO - 2;
    int lz = (int)p[2]; if (lz > RESO - 2) lz = RESO - 2;

    const float wbx = p[0] - (float)lx, wax = 1.0f - wbx;
    const float wby = p[1] - (float)ly, way = 1.0f - wby;
    const float wbz = p[2] - (float)lz, waz = 1.0f - wbz;
    const float wx[2] = {wax, wbx};
    const float wy[2] = {way, wby};
    const float wz[2] = {waz, wbz};

    const int base = (lx * RESO + ly) * RESO + lz;

    // Hoist all 8 link gathers: z/z+1 pairs are adjacent ints (b64-mergeable),
    // and 8 loads go in flight together.
    int lk[8];
#pragma unroll
    for (int c = 0; c < 8; ++c) {
        const int dx = (c >> 2) & 1, dy = (c >> 1) & 1, dz = c & 1;
        lk[c] = links[base + dx * RESO * RESO + dy * RESO + dz];
    }

    f4 acc[DATA_F4];
#pragma unroll
    for (int j = 0; j < DATA_F4; ++j) acc[j] = (f4){0.0f, 0.0f, 0.0f, 0.0f};

#pragma unroll
    for (int c = 0; c < 8; ++c) {
        if (lk[c] >= 0) {                        // skip empty voxels (~30%): EXEC-masked loads
            const int dx = (c >> 2) & 1, dy = (c >> 1) & 1, dz = c & 1;
            const float w = wx[dx] * wy[dy] * wz[dz];
            const f4* row = data + (long long)lk[c] * DATA_F4;  // 16B-aligned, 7 x b128
#pragma unroll
            for (int j = 0; j < DATA_F4; ++j) {
                f4 v = row[j];
                acc[j] += w * v;
            }
        }
    }

    // Streaming (non-temporal) b128 stores: output is write-once; keep L2 for `data`.
    f4* orow = out + (long long)i * DATA_F4;
#pragma unroll
    for (int j = 0; j < DATA_F4; ++j)
        __builtin_nontemporal_store(acc[j], orow + j);
}

extern "C" void kernel_launch(void* const* d_in, const int* in_sizes, int n_in,
                              void* d_out, int out_size, void* d_ws, size_t ws_size,
                              hipStream_t stream) {
    const float* points = (const float*)d_in[0];
    const f4*    data   = (const f4*)d_in[1];
    const int*   links  = (const int*)d_in[2];
    const float* offs   = (const float*)d_in[3];
    const float* scal   = (const float*)d_in[4];
    f4*          out    = (f4*)d_out;

    const int npts   = in_sizes[0] / 3;
    const int blocks = (npts + BLOCK - 1) / BLOCK;
    trilerp_sparse_kernel<<<blocks, BLOCK, 0, stream>>>(points, data, links,
                                                        offs, scal, out, npts);
}